// FlatContinuousDecoder_63316407878210
// MI455X (gfx1250) — compile-verified
//
#include <hip/hip_runtime.h>
#include <math.h>

// Problem constants (match the reference)
constexpr int VSZ = 32000;   // vocab
constexpr int KD  = 32;      // latent dim
constexpr int RR  = 8;       // routes
constexpr int BB  = 1024;    // batch
constexpr int NTILE = 128;   // vocab columns per block (8 waves * 16)
constexpr int MAXG  = 64;    // max 16-sample groups per route (ceil(1024/16))

typedef __attribute__((ext_vector_type(2))) float v2f;
typedef __attribute__((ext_vector_type(8))) float v8f;

// ---------------------------------------------------------------------------
// Pass 1: bucket samples by route so each WMMA tile shares one eta_axes slice.
// ---------------------------------------------------------------------------
__global__ __launch_bounds__(1024)
void compact_routes(const int* __restrict__ r_idx,
                    int* __restrict__ counts,
                    int* __restrict__ sorted)
{
    __shared__ int lcnt[RR];
    const int t = threadIdx.x;
    if (t < RR) lcnt[t] = 0;
    __syncthreads();
    if (t < BB) {
        int r = r_idx[t];
        int pos = atomicAdd(&lcnt[r], 1);
        sorted[r * BB + pos] = t;
    }
    __syncthreads();
    if (t < RR) counts[t] = lcnt[t];
}

// ---------------------------------------------------------------------------
// Pass 2: routed GEMM via V_WMMA_F32_16X16X4_F32 + fused bias/author gather.
// grid = (VSZ/NTILE, RR*MAXG), block = 256 (8 wave32).
// ---------------------------------------------------------------------------
__global__ __launch_bounds__(256)
void routed_gemm(const int*   __restrict__ m_idx,
                 const float* __restrict__ z,
                 const float* __restrict__ eta_bg,
                 const float* __restrict__ eta_author,
                 const float* __restrict__ eta_axes,
                 const int*   __restrict__ counts,
                 const int*   __restrict__ sorted,
                 float*       __restrict__ out)
{
    const int r = blockIdx.y >> 6;      // route
    const int g = blockIdx.y & 63;      // 16-sample group within route
    const int cnt = counts[r];
    if (g * 16 >= cnt) return;          // uniform per block -> EXEC stays full

    __shared__ float z_lds[16][KD];
    __shared__ int   b_sel[16];
    __shared__ int   m_sel[16];

    const int tid = threadIdx.x;
    if (tid < 16) {
        int s = g * 16 + tid;
        int b = (s < cnt) ? sorted[r * BB + s] : -1;
        b_sel[tid] = b;
        m_sel[tid] = (b >= 0) ? m_idx[b] : 0;
    }
    __syncthreads();
    // Cooperative load of the 16 z rows (512 floats); zero-pad invalid rows.
    for (int i = tid; i < 16 * KD; i += 256) {
        int row = i >> 5, k = i & 31;
        int b = b_sel[row];
        z_lds[row][k] = (b >= 0) ? z[b * KD + k] : 0.0f;
    }
    __syncthreads();

    const int wave = tid >> 5;
    const int lane = tid & 31;
    const int mrow = lane & 15;              // A: M index / B,C: N index
    const int kb   = (lane >> 4) << 1;       // 0 (lanes 0-15) or 2 (lanes 16-31)
    const int vcol = blockIdx.x * NTILE + wave * 16 + mrow;

    v8f c = {};
    // eta_axes layout: [K][R][V] -> element (k, r, v) at ((k*RR + r)*VSZ + v)
#pragma unroll
    for (int kk = 0; kk < 8; ++kk) {
        const int k0 = kk * 4 + kb;
        v2f a, bm;
        a.x  = z_lds[mrow][k0];
        a.y  = z_lds[mrow][k0 + 1];
        bm.x = eta_axes[((size_t)(k0)     * RR + r) * VSZ + vcol];
        bm.y = eta_axes[((size_t)(k0 + 1) * RR + r) * VSZ + vcol];
        // D = A(16x4,f32) * B(4x16,f32) + C ; 8-arg form per probe doc
        c = __builtin_amdgcn_wmma_f32_16x16x4_f32(
                /*neg_a=*/false, a, /*neg_b=*/false, bm,
                /*c_mod=*/(short)0, c, /*reuse_a=*/false, /*reuse_b=*/false);
    }

    const float bgv  = eta_bg[(size_t)r * VSZ + vcol];
    const int   roff = (lane >> 4) << 3;     // C rows: vgpr j -> M = j + roff
#pragma unroll
    for (int j = 0; j < 8; ++j) {
        int row = j + roff;
        int b = b_sel[row];
        if (b >= 0) {
            float val = c[j] + bgv +
                eta_author[((size_t)m_sel[row] * RR + r) * VSZ + vcol];
            out[(size_t)b * VSZ + vcol] = val;
        }
    }
}

// ---------------------------------------------------------------------------
// Pass 3: per-row online log-sum-exp. grid = B, block = 256.
// ---------------------------------------------------------------------------
__global__ __launch_bounds__(256)
void row_logsumexp(const float* __restrict__ out, float* __restrict__ shift)
{
    const int b = blockIdx.x;
    const float* row = out + (size_t)b * VSZ;
    float m = -INFINITY, s = 0.0f;
    for (int v = threadIdx.x; v < VSZ; v += 256) {
        float x = row[v];
        if (x > m) { s *= __expf(m - x); m = x; }
        s += __expf(x - m);
    }
    __shared__ float sm[256];
    __shared__ float ss[256];
    const int t = threadIdx.x;
    sm[t] = m; ss[t] = s;
    __syncthreads();
    for (int off = 128; off > 0; off >>= 1) {
        if (t < off) {
            float m2 = sm[t + off], s2 = ss[t + off];
            float mm = fmaxf(sm[t], m2);
            ss[t] = ss[t] * __expf(sm[t] - mm) + s2 * __expf(m2 - mm);
            sm[t] = mm;
        }
        __syncthreads();
    }
    if (t == 0) shift[b] = sm[0] + __logf(ss[0]);
}

// ---------------------------------------------------------------------------
// Pass 4: out[b,v] -= shift[b].  grid = (VSZ/256, B), block = 256.
// ---------------------------------------------------------------------------
__global__ __launch_bounds__(256)
void apply_shift(float* __restrict__ out, const float* __restrict__ shift)
{
    const int b = blockIdx.y;
    const int v = blockIdx.x * 256 + threadIdx.x;
    const float sh = shift[b];
    out[(size_t)b * VSZ + v] -= sh;
}

// ---------------------------------------------------------------------------
extern "C" void kernel_launch(void* const* d_in, const int* in_sizes, int n_in,
                              void* d_out, int out_size, void* d_ws, size_t ws_size,
                              hipStream_t stream)
{
    const int*   m_idx      = (const int*)  d_in[0];
    const float* z          = (const float*)d_in[1];
    const int*   r_idx      = (const int*)  d_in[2];
    const float* eta_bg     = (const float*)d_in[3];
    const float* eta_author = (const float*)d_in[4];
    const float* eta_axes   = (const float*)d_in[5];
    float* out = (float*)d_out;

    // Workspace layout: counts[R] | sorted[R*B] | shift[B]
    char* ws = (char*)d_ws;
    int*   counts = (int*)ws;
    int*   sorted = (int*)(ws + 64);
    float* shift  = (float*)(ws + 64 + (size_t)RR * BB * sizeof(int));

    hipLaunchKernelGGL(compact_routes, dim3(1), dim3(1024), 0, stream,
                       r_idx, counts, sorted);
    hipLaunchKernelGGL(routed_gemm, dim3(VSZ / NTILE, RR * MAXG), dim3(256), 0, stream,
                       m_idx, z, eta_bg, eta_author, eta_axes, counts, sorted, out);
    hipLaunchKernelGGL(row_logsumexp, dim3(BB), dim3(256), 0, stream, out, shift);
    hipLaunchKernelGGL(apply_shift, dim3(VSZ / 256, BB), dim3(256), 0, stream,
                       out, shift);
}